// GNN_52390011076772
// MI455X (gfx1250) — compile-verified
//
#include <hip/hip_runtime.h>

#define D 128
#define LDS_PAD 130  // stride-130 padding: lanes 0-15 hit banks 0,2,4,... -> conflict-free

typedef __attribute__((ext_vector_type(2))) float v2f;
typedef __attribute__((ext_vector_type(8))) float v8f;

// ---------------------------------------------------------------------------
// Zero-fill scratch (agg [+cnt]) — harness poisons ws, we must re-zero per call.
// ---------------------------------------------------------------------------
__global__ void GNN_zero_kernel(float* __restrict__ p, int n) {
    int i = blockIdx.x * blockDim.x + threadIdx.x;
    if (i < n) p[i] = 0.0f;
}

// ---------------------------------------------------------------------------
// Edge scatter: one wave32 per edge. Lane l moves 4 consecutive floats of the
// 128-float source row (float4 load = 512B/row fully coalesced per wave) and
// atomically accumulates into agg[dst]. Lane 0 bumps the degree counter
// (only on layer 1 — the degree is identical for layer 2, so we reuse it).
// ---------------------------------------------------------------------------
__global__ __launch_bounds__(256) void GNN_scatter_kernel(
    const float* __restrict__ x,
    const long long* __restrict__ src,
    const long long* __restrict__ dst,
    float* __restrict__ agg,
    float* __restrict__ cnt,
    int E)
{
    int gid  = blockIdx.x * blockDim.x + threadIdx.x;
    int e    = gid >> 5;
    int lane = gid & 31;
    if (e >= E) return;

    int s = (int)src[e];
    int d = (int)dst[e];

    const float4 v = ((const float4*)(x + (size_t)s * D))[lane];
    float* a = agg + (size_t)d * D + lane * 4;
    (void)__hip_atomic_fetch_add(a + 0, v.x, __ATOMIC_RELAXED, __HIP_MEMORY_SCOPE_AGENT);
    (void)__hip_atomic_fetch_add(a + 1, v.y, __ATOMIC_RELAXED, __HIP_MEMORY_SCOPE_AGENT);
    (void)__hip_atomic_fetch_add(a + 2, v.z, __ATOMIC_RELAXED, __HIP_MEMORY_SCOPE_AGENT);
    (void)__hip_atomic_fetch_add(a + 3, v.w, __ATOMIC_RELAXED, __HIP_MEMORY_SCOPE_AGENT);
    if (cnt != nullptr && lane == 0)
        (void)__hip_atomic_fetch_add(cnt + d, 1.0f, __ATOMIC_RELAXED, __HIP_MEMORY_SCOPE_AGENT);
}

// ---------------------------------------------------------------------------
// Fused SAGE layer: out[16 nodes] = mean@Wl + bl + x@Wr (+ReLU).
// 256 threads = 8 waves; wave w owns output columns [16w, 16w+16).
// A tiles (mean, x) staged in LDS; W stripes read from global (L2-resident).
// f32 WMMA 16x16x4: A 16x4 (v2f), B 4x16 (v2f), C/D 16x16 (v8f).
//   A layout: lane = (k>=2 ? 16:0) + m, vgpr = k&1        -> a[v] = A[m][k0+v]
//   B layout: lane = (k>=2 ? 16:0) + n, vgpr = k&1        -> b[v] = B[k0+v][n]
//   C layout: row = v + 8*(lane>>4), col = lane&15
// ---------------------------------------------------------------------------
__global__ __launch_bounds__(256) void GNN_sage_layer_kernel(
    const float* __restrict__ xin,  // [N,128] layer input
    const float* __restrict__ agg,  // [N,128] summed neighbor features
    const float* __restrict__ cnt,  // [N]     in-degree (float)
    const float* __restrict__ Wl,   // [128,128]
    const float* __restrict__ bl,   // [128]
    const float* __restrict__ Wr,   // [128,128]
    float* __restrict__ out,        // [N,128]
    int relu, int N)
{
    __shared__ float sMean[16 * LDS_PAD];
    __shared__ float sX[16 * LDS_PAD];

    const int t = threadIdx.x;
    const int nodeBase = blockIdx.x * 16;

    // Cooperative stage: 2048 elems per tile, 8 per thread, coalesced.
#pragma unroll
    for (int i = 0; i < 8; ++i) {
        int e   = t + 256 * i;
        int row = e >> 7;
        int col = e & (D - 1);
        int r   = nodeBase + row;
        float mv = 0.0f, xv = 0.0f;
        if (r < N) {
            float c  = cnt[r];
            float inv = 1.0f / fmaxf(c, 1.0f);
            mv = agg[(size_t)r * D + col] * inv;
            xv = xin[(size_t)r * D + col];
        }
        sMean[row * LDS_PAD + col] = mv;
        sX[row * LDS_PAD + col]    = xv;
    }
    __syncthreads();

    const int wave  = t >> 5;
    const int lane  = t & 31;
    const int m     = lane & 15;   // A row / B-C column within tile
    const int khalf = lane >> 4;   // 0 or 1
    const int nb    = wave * 16;   // output column base

    v8f acc = {};

    // mean @ Wl
#pragma unroll
    for (int kk = 0; kk < D; kk += 4) {
        const int ka = kk + khalf * 2;               // even -> v2f aligned
        v2f a = *(const v2f*)&sMean[m * LDS_PAD + ka];
        v2f b;
        b.x = Wl[(size_t)ka * D + nb + m];
        b.y = Wl[(size_t)(ka + 1) * D + nb + m];
        acc = __builtin_amdgcn_wmma_f32_16x16x4_f32(false, a, false, b,
                                                    (short)0, acc, false, false);
    }
    // x @ Wr (accumulate into same C)
#pragma unroll
    for (int kk = 0; kk < D; kk += 4) {
        const int ka = kk + khalf * 2;
        v2f a = *(const v2f*)&sX[m * LDS_PAD + ka];
        v2f b;
        b.x = Wr[(size_t)ka * D + nb + m];
        b.y = Wr[(size_t)(ka + 1) * D + nb + m];
        acc = __builtin_amdgcn_wmma_f32_16x16x4_f32(false, a, false, b,
                                                    (short)0, acc, false, false);
    }

    const float bias = bl[nb + m];
#pragma unroll
    for (int v = 0; v < 8; ++v) {
        int r = nodeBase + v + 8 * khalf;
        if (r < N) {
            float o = acc[v] + bias;
            if (relu) o = fmaxf(o, 0.0f);
            out[(size_t)r * D + nb + m] = o;
        }
    }
}

// ---------------------------------------------------------------------------
// Inputs (setup_inputs order):
//   0: x [N,128] f32     1: edge_index [2,E] i64
//   2: W_l1 [128,128]    3: b_l1 [128]    4: W_r1 [128,128]
//   5: W_l2 [128,128]    6: b_l2 [128]    7: W_r2 [128,128]
// Output: [N,128] f32.
// Workspace: agg (N*128) | cnt (N) | h (N*128)  -> ~103 MB f32.
// ---------------------------------------------------------------------------
extern "C" void kernel_launch(void* const* d_in, const int* in_sizes, int n_in,
                              void* d_out, int out_size, void* d_ws, size_t ws_size,
                              hipStream_t stream) {
    const float*     x   = (const float*)d_in[0];
    const long long* ei  = (const long long*)d_in[1];
    const float*     Wl1 = (const float*)d_in[2];
    const float*     bl1 = (const float*)d_in[3];
    const float*     Wr1 = (const float*)d_in[4];
    const float*     Wl2 = (const float*)d_in[5];
    const float*     bl2 = (const float*)d_in[6];
    const float*     Wr2 = (const float*)d_in[7];

    const int N = in_sizes[0] / D;
    const int E = in_sizes[1] / 2;
    const long long* src = ei;
    const long long* dst = ei + E;

    float* agg = (float*)d_ws;
    float* cnt = agg + (size_t)N * D;
    float* h   = cnt + N;
    float* out = (float*)d_out;

    const int nBlocksGemm    = (N + 15) / 16;
    const int nScatterBlocks = (int)(((long long)E * 32 + 255) / 256);

    // ---- Layer 1 ----
    const int nZero1 = N * D + N;  // agg + cnt contiguous
    GNN_zero_kernel<<<(nZero1 + 255) / 256, 256, 0, stream>>>(agg, nZero1);
    GNN_scatter_kernel<<<nScatterBlocks, 256, 0, stream>>>(x, src, dst, agg, cnt, E);
    GNN_sage_layer_kernel<<<nBlocksGemm, 256, 0, stream>>>(x, agg, cnt, Wl1, bl1, Wr1,
                                                           h, /*relu=*/1, N);
    // ---- Layer 2 (reuse cnt: degrees identical) ----
    GNN_zero_kernel<<<(N * D + 255) / 256, 256, 0, stream>>>(agg, N * D);
    GNN_scatter_kernel<<<nScatterBlocks, 256, 0, stream>>>(h, src, dst, agg, nullptr, E);
    GNN_sage_layer_kernel<<<nBlocksGemm, 256, 0, stream>>>(h, agg, cnt, Wl2, bl2, Wr2,
                                                           out, /*relu=*/0, N);
}